// Relation_Module_45664092291701
// MI455X (gfx1250) — compile-verified
//
#include <hip/hip_runtime.h>

// Relation module on MI455X (gfx1250, wave32).
// All GEMM-shaped stages run on V_WMMA_F32_16X16X4_F32 (full fp32 fidelity).
// Big GEMMs use 64x64-per-wave register blocking: 4x4 grid of 16x16 WMMA
// accumulators -> 16 independent back-to-back WMMAs per K-step and ~4x less
// L2 traffic than 16x16 tiling (AI ~16 flop/byte vs ~4).

typedef float v2f __attribute__((ext_vector_type(2)));
typedef float v8f __attribute__((ext_vector_type(8)));

#define RN      1024
#define RDIM_V  2048
#define RDIM_MM 64
#define RDIM_O  2048

// ---------------------------------------------------------------------------
// 64x64 output block per wave.
// MODE 0: C = relu(A) @ B + bias      (aux = bias[Ncols])
// MODE 1: C = (A @ B) * aux[row]      (aux = reciprocal row sums)
// A: [M, Kdim] row-major, B: [Kdim, Ncols] row-major, C: [M, Ncols].
// f32 WMMA lane layout: A frag VGPR r -> A[M=lm, K=2g+r];
//                       B frag VGPR r -> B[K=2g+r, N=lm];
//                       C       VGPR r -> C[M=r+8g, N=lm].
// ---------------------------------------------------------------------------
template<int MODE>
__global__ __launch_bounds__(32) void gemm64_kernel(
    const float* __restrict__ A, const float* __restrict__ B,
    const float* __restrict__ aux, float* __restrict__ C,
    int Kdim, int Ncols)
{
    const int lane = threadIdx.x;        // 0..31
    const int g    = lane >> 4;          // lane group 0/1
    const int lm   = lane & 15;
    const int row0 = blockIdx.x * 64;
    const int col0 = blockIdx.y * 64;

    v8f c[4][4] = {};

    for (int kk = 0; kk < Kdim; kk += 4) {
        // Prefetch the B panel 4 K-steps ahead (uniform branch, no divergence).
        if (kk + 16 < Kdim) {
            __builtin_prefetch(B + (size_t)(kk + 16) * Ncols + col0 + 2 * lane, 0, 1);
        }

        v2f a[4];
        #pragma unroll
        for (int mi = 0; mi < 4; ++mi) {
            const float* ap = A + (size_t)(row0 + 16 * mi + lm) * Kdim + kk + 2 * g;
            float x = ap[0], y = ap[1];
            if (MODE == 0) { x = fmaxf(x, 0.0f); y = fmaxf(y, 0.0f); }
            a[mi].x = x; a[mi].y = y;
        }
        v2f b[4];
        #pragma unroll
        for (int ni = 0; ni < 4; ++ni) {
            const float* bp = B + (size_t)(kk + 2 * g) * Ncols + col0 + 16 * ni + lm;
            b[ni].x = bp[0];
            b[ni].y = bp[Ncols];
        }
        // 16 independent WMMAs: no accumulator RAW hazards within a K-step.
        #pragma unroll
        for (int mi = 0; mi < 4; ++mi)
            #pragma unroll
            for (int ni = 0; ni < 4; ++ni)
                c[mi][ni] = __builtin_amdgcn_wmma_f32_16x16x4_f32(
                    false, a[mi], false, b[ni], (short)0, c[mi][ni], false, false);
    }

    #pragma unroll
    for (int mi = 0; mi < 4; ++mi) {
        #pragma unroll
        for (int r = 0; r < 8; ++r) {
            const int m = row0 + 16 * mi + r + 8 * g;
            const float rscale = (MODE == 1) ? aux[m] : 0.0f;
            #pragma unroll
            for (int ni = 0; ni < 4; ++ni) {
                const int n = col0 + 16 * ni + lm;
                float val = c[mi][ni][r];
                if (MODE == 0) val += aux[n];
                else           val *= rscale;
                C[(size_t)m * Ncols + n] = val;
            }
        }
    }
}

// ---------------------------------------------------------------------------
// S[i,j] = (i==j) ? 0 : exp((k_i . q_j) / 8) * geoMLP(roi_i, roi_j)
// k q^T on WMMA (K=64); geometry 4->64->1 MLP on VALU per element.
// One wave per 16x16 tile of S.
// ---------------------------------------------------------------------------
__global__ __launch_bounds__(32) void score_kernel(
    const float* __restrict__ kmat, const float* __restrict__ qmat,
    const float* __restrict__ rois,
    const float* __restrict__ G1, const float* __restrict__ g1,
    const float* __restrict__ G2, const float* __restrict__ g2,
    float* __restrict__ S)
{
    const int lane = threadIdx.x;
    const int g    = lane >> 4;
    const int lm   = lane & 15;
    const int row0 = blockIdx.x * 16;    // i tile
    const int col0 = blockIdx.y * 16;    // j tile

    // vis[i,j] = sum_f k[i,f] * q[j,f]  ->  A = k tile, B = q^T tile
    v8f c = {};
    #pragma unroll
    for (int kk = 0; kk < RDIM_MM; kk += 4) {
        v2f a, b;
        a.x = kmat[(row0 + lm) * RDIM_MM + kk + 2 * g + 0];
        a.y = kmat[(row0 + lm) * RDIM_MM + kk + 2 * g + 1];
        b.x = qmat[(col0 + lm) * RDIM_MM + kk + 2 * g + 0];
        b.y = qmat[(col0 + lm) * RDIM_MM + kk + 2 * g + 1];
        c = __builtin_amdgcn_wmma_f32_16x16x4_f32(
                false, a, false, b, (short)0, c, false, false);
    }

    // Geometry for column j (fixed per lane)
    const int j = col0 + lm;
    const float jx1 = rois[j * 4 + 0], jy1 = rois[j * 4 + 1];
    const float jx2 = rois[j * 4 + 2], jy2 = rois[j * 4 + 3];
    const float wj = jx2 - jx1, hj = jy2 - jy1;
    const float cjx = 0.5f * (jx1 + jx2), cjy = 0.5f * (jy1 + jy2);
    const float g2s = g2[0];

    #pragma unroll
    for (int r = 0; r < 8; ++r) {
        const int i = row0 + r + 8 * g;
        const float ix1 = rois[i * 4 + 0], iy1 = rois[i * 4 + 1];
        const float ix2 = rois[i * 4 + 2], iy2 = rois[i * 4 + 3];
        const float wi = ix2 - ix1, hi = iy2 - iy1;
        const float cix = 0.5f * (ix1 + ix2), ciy = 0.5f * (iy1 + iy2);
        const float dx = cjx - cix, dy = cjy - ciy;
        const float f1 = logf(fabsf(dx) / wi + 1e-3f);
        const float f2 = logf(fabsf(dy) / hi + 1e-3f);
        const float f3 = logf(wj / wi);
        const float f4 = logf(hj / hi);
        // h = relu(geo @ G1 + g1); w = relu(h @ G2 + g2)
        float acc = g2s;
        for (int m = 0; m < RDIM_MM; ++m) {
            float h = f1 * G1[0 * RDIM_MM + m] + f2 * G1[1 * RDIM_MM + m]
                    + f3 * G1[2 * RDIM_MM + m] + f4 * G1[3 * RDIM_MM + m]
                    + g1[m];
            acc += fmaxf(h, 0.0f) * G2[m];
        }
        const float geow = fmaxf(acc, 0.0f);
        const float vis  = c[r] * 0.125f;           // 1/sqrt(64)
        const float s    = (i == j) ? 0.0f : expf(vis) * geow;
        S[(size_t)i * RN + j] = s;
    }
}

// ---------------------------------------------------------------------------
// rs[i] = 1 / (sum_j S[i,j] + 1e-10)
// ---------------------------------------------------------------------------
__global__ __launch_bounds__(256) void rowsum_kernel(
    const float* __restrict__ S, float* __restrict__ rs)
{
    __shared__ float sm[256];
    const int i = blockIdx.x;
    float acc = 0.0f;
    for (int j = threadIdx.x; j < RN; j += 256)
        acc += S[(size_t)i * RN + j];
    sm[threadIdx.x] = acc;
    __syncthreads();
    for (int s = 128; s > 0; s >>= 1) {
        if (threadIdx.x < s) sm[threadIdx.x] += sm[threadIdx.x + s];
        __syncthreads();
    }
    if (threadIdx.x == 0) rs[i] = 1.0f / (sm[0] + 1e-10f);
}

// ---------------------------------------------------------------------------
extern "C" void kernel_launch(void* const* d_in, const int* in_sizes, int n_in,
                              void* d_out, int out_size, void* d_ws, size_t ws_size,
                              hipStream_t stream) {
    const float* fv   = (const float*)d_in[0];   // [1024, 2048]
    const float* rois = (const float*)d_in[1];   // [1024, 4]
    const float* Wk   = (const float*)d_in[2];   // [2048, 64]
    const float* bk   = (const float*)d_in[3];   // [64]
    const float* Wq   = (const float*)d_in[4];   // [2048, 64]
    const float* bq   = (const float*)d_in[5];   // [64]
    const float* Wv   = (const float*)d_in[6];   // [2048, 2048]
    const float* bv   = (const float*)d_in[7];   // [2048]
    const float* G1   = (const float*)d_in[8];   // [4, 64]
    const float* g1   = (const float*)d_in[9];   // [64]
    const float* G2   = (const float*)d_in[10];  // [64, 1]
    const float* g2   = (const float*)d_in[11];  // [1]
    float* out = (float*)d_out;                  // [1024, 2048]

    // Workspace layout (floats): k, q, v, S, recip-rowsum  (~13.1 MB total)
    float* ws  = (float*)d_ws;
    float* k_  = ws;                                        // 1024*64
    float* q_  = k_ + (size_t)RN * RDIM_MM;                 // 1024*64
    float* v_  = q_ + (size_t)RN * RDIM_MM;                 // 1024*2048
    float* S_  = v_ + (size_t)RN * RDIM_O;                  // 1024*1024
    float* rs_ = S_ + (size_t)RN * RN;                      // 1024

    const dim3 wave(32);

    // k, q projections: [1024x2048] @ [2048x64] (one 64-wide column group)
    gemm64_kernel<0><<<dim3(RN / 64, RDIM_MM / 64), wave, 0, stream>>>(
        fv, Wk, bk, k_, RDIM_V, RDIM_MM);
    gemm64_kernel<0><<<dim3(RN / 64, RDIM_MM / 64), wave, 0, stream>>>(
        fv, Wq, bq, q_, RDIM_V, RDIM_MM);
    // v projection: [1024x2048] @ [2048x2048]
    gemm64_kernel<0><<<dim3(RN / 64, RDIM_O / 64), wave, 0, stream>>>(
        fv, Wv, bv, v_, RDIM_V, RDIM_O);
    // scores: WMMA k q^T + fused geometry MLP + exp + zero diagonal
    score_kernel<<<dim3(RN / 16, RN / 16), wave, 0, stream>>>(
        k_, q_, rois, G1, g1, G2, g2, S_);
    // per-row reciprocal sums
    rowsum_kernel<<<dim3(RN), dim3(256), 0, stream>>>(S_, rs_);
    // out = (S @ v) scaled by rs (row normalization folded into epilogue)
    gemm64_kernel<1><<<dim3(RN / 64, RDIM_O / 64), wave, 0, stream>>>(
        S_, v_, rs_, out, RN, RDIM_O);
}